// HiddenMarkovStateInferenceLayer_15101105013083
// MI455X (gfx1250) — compile-verified
//
#include <hip/hip_runtime.h>

// Baum-Welch forward/backward for N=12-state HMM, T=4096, BATCH=64.
// fp32 throughout (matches reference numerics). Per-step recurrences are
// (16x12)@(12x12) GEMMs done with V_WMMA_F32_16X16X4_F32 (K=12 -> 3 chunks).
//
// Workspace (floats), ~26.2 MB total (L2-resident on MI455X's 192MB L2):
//   ws_alpha : T*BATCH*NS   time-major [t][b][n]
//   ws_scale : T*BATCH
//   ws_beta  : T*BATCH*NS

#define T_LEN 4096
#define BATCH 64
#define NS    12
#define EPS_F 1.1920929e-07f   // FLT_EPSILON

typedef __attribute__((ext_vector_type(2))) float v2f;
typedef __attribute__((ext_vector_type(8))) float v8f;

// 12-float row helpers (rows are 48B and 16B-aligned everywhere we use them).
__device__ __forceinline__ void load_row12(const float* __restrict__ p, float v[NS]) {
    const float4 q0 = *reinterpret_cast<const float4*>(p);
    const float4 q1 = *reinterpret_cast<const float4*>(p + 4);
    const float4 q2 = *reinterpret_cast<const float4*>(p + 8);
    v[0]=q0.x; v[1]=q0.y; v[2]=q0.z; v[3]=q0.w;
    v[4]=q1.x; v[5]=q1.y; v[6]=q1.z; v[7]=q1.w;
    v[8]=q2.x; v[9]=q2.y; v[10]=q2.z; v[11]=q2.w;
}
__device__ __forceinline__ void store_row12(float* __restrict__ p, const float v[NS]) {
    *reinterpret_cast<float4*>(p)     = make_float4(v[0], v[1], v[2],  v[3]);
    *reinterpret_cast<float4*>(p + 4) = make_float4(v[4], v[5], v[6],  v[7]);
    *reinterpret_cast<float4*>(p + 8) = make_float4(v[8], v[9], v[10], v[11]);
}
__device__ __forceinline__ float rowmax_shift(const float lb[NS]) {
    float m = lb[0];
#pragma unroll
    for (int j = 1; j < NS; ++j) m = fmaxf(m, lb[j]);
    return fminf(m, 0.0f);
}

// -------------------------------------------------------------------------
// Forward scan: alpha_t = normalize((alpha_{t-1} @ P) * B_t), 16-batch tile
// per single-wave block. Fragment layouts (ISA 7.12.2, 32-bit, wave32):
//   A 16x4 : lane<16 -> row M=lane, vgpr{0,1}=K{0,1}; lane>=16 -> K{2,3}
//   B 4x16 : vgpr0: K = 2*half, vgpr1: K = 2*half+1, N = lane&15
//   C/D    : vgpr v: M = v + 8*half, N = lane&15
// -------------------------------------------------------------------------
__global__ __launch_bounds__(32)
void hmm_forward_scan(const float* __restrict__ logB,
                      const float* __restrict__ Pi0,
                      const float* __restrict__ P,
                      float* __restrict__ ws_alpha,
                      float* __restrict__ ws_scale)
{
    __shared__ __align__(16) float lds_a[16][16];  // row-major normalized alpha_{t-1}
    __shared__ __align__(16) float prodT[16][16];  // [state][batch_row] raw WMMA out

    const int lane = threadIdx.x;
    const int half = lane >> 4;
    const int l    = lane & 15;
    const int b0   = blockIdx.x * 16;

    // Resident B-fragments of P (K-chunks = rows 4c..4c+3 of P).
    v2f pfrag[3];
#pragma unroll
    for (int c = 0; c < 3; ++c) {
        const int k0 = 4 * c + 2 * half;              // <= 11
        pfrag[c].x = (l < NS) ? P[k0 * NS + l]       : 0.0f;
        pfrag[c].y = (l < NS) ? P[(k0 + 1) * NS + l] : 0.0f;
    }

    // t = 0: a0 = Pi0 * B0, rescaled. Row-owner lanes 0-15 only.
    if (lane < 16) {
        float lb[NS];
        load_row12(logB + (size_t)(b0 + l) * T_LEN * NS, lb);
        const float m = rowmax_shift(lb);
        float a[NS]; float s0 = 0.0f;
#pragma unroll
        for (int j = 0; j < NS; ++j) { a[j] = Pi0[j] * __expf(lb[j] - m); s0 += a[j]; }
        const float inv = 1.0f / (s0 + EPS_F);
#pragma unroll
        for (int j = 0; j < NS; ++j) a[j] *= inv;
        store_row12(&lds_a[l][0], a);
        store_row12(ws_alpha + (size_t)(b0 + l) * NS, a);
        ws_scale[b0 + l] = s0;
    }
    __syncthreads();

    for (int t = 1; t < T_LEN; ++t) {
        // Row-owner: B_t row in registers only (never through LDS).
        float Bv[NS];
        if (lane < 16) {
            const float* row = logB + ((size_t)(b0 + l) * T_LEN + t) * NS;
            float lb[NS];
            load_row12(row, lb);
            const float m = rowmax_shift(lb);
#pragma unroll
            for (int j = 0; j < NS; ++j) Bv[j] = __expf(lb[j] - m);
            if (t + 1 < T_LEN) __builtin_prefetch(row + NS, 0, 3);
        }

        // alpha_{t-1} @ P : 3 accumulating K=4 WMMAs.
        v8f acc = {};
#pragma unroll
        for (int c = 0; c < 3; ++c) {
            const int k = 4 * c + 2 * half;
            v2f a;
            a.x = lds_a[l][k];
            a.y = lds_a[l][k + 1];
            acc = __builtin_amdgcn_wmma_f32_16x16x4_f32(false, a, false, pfrag[c],
                                                        (short)0, acc, false, false);
        }
        // C-layout -> transposed tile: 2x ds_store_b128 per lane.
        {
            float4* dst = reinterpret_cast<float4*>(&prodT[l][8 * half]);
            dst[0] = make_float4(acc[0], acc[1], acc[2], acc[3]);
            dst[1] = make_float4(acc[4], acc[5], acc[6], acc[7]);
        }
        __syncthreads();   // prodT visible; lds_a reads done before overwrite

        // Row-owner: * B_t, row-sum, rescale, write back + coalesced global row.
        if (lane < 16) {
            float p[NS]; float s = 0.0f;
#pragma unroll
            for (int j = 0; j < NS; ++j) { p[j] = prodT[j][l] * Bv[j]; s += p[j]; }
            ws_scale[(size_t)t * BATCH + b0 + l] = s;
            const float inv = 1.0f / (s + EPS_F);
            float a[NS];
#pragma unroll
            for (int j = 0; j < NS; ++j) a[j] = p[j] * inv;
            store_row12(&lds_a[l][0], a);
            store_row12(ws_alpha + ((size_t)t * BATCH + b0 + l) * NS, a);
        }
        __syncthreads();   // lds_a visible; prodT reads done before next stores
    }
}

// -------------------------------------------------------------------------
// Backward scan: beta_t = (x_{t+1} @ P^T) / (scale_t + eps),
// x_{t+1} = beta_{t+1} .* B_{t+1} (formed in row-owner registers).
// -------------------------------------------------------------------------
__global__ __launch_bounds__(32)
void hmm_backward_scan(const float* __restrict__ logB,
                       const float* __restrict__ P,
                       const float* __restrict__ ws_scale,
                       float* __restrict__ ws_beta)
{
    __shared__ __align__(16) float lds_x[16][16];  // row-major x_{t+1}
    __shared__ __align__(16) float prodT[16][16];  // [state][batch_row]

    const int lane = threadIdx.x;
    const int half = lane >> 4;
    const int l    = lane & 15;
    const int b0   = blockIdx.x * 16;

    // B-fragments of P^T: element (K=j, N=k) = P[k][j].
    v2f ptfrag[3];
#pragma unroll
    for (int c = 0; c < 3; ++c) {
        const int k0 = 4 * c + 2 * half;
        ptfrag[c].x = (l < NS) ? P[l * NS + k0]     : 0.0f;
        ptfrag[c].y = (l < NS) ? P[l * NS + k0 + 1] : 0.0f;
    }

    // t = T-1: beta = 1/(scale+eps); seed x_{T-1} = beta .* B_{T-1}.
    if (lane < 16) {
        const float s  = ws_scale[(size_t)(T_LEN - 1) * BATCH + b0 + l];
        const float bv = 1.0f / (s + EPS_F);
        float r[NS];
#pragma unroll
        for (int j = 0; j < NS; ++j) r[j] = bv;
        store_row12(ws_beta + ((size_t)(T_LEN - 1) * BATCH + b0 + l) * NS, r);
        float lb[NS];
        load_row12(logB + ((size_t)(b0 + l) * T_LEN + (T_LEN - 1)) * NS, lb);
        const float m = rowmax_shift(lb);
        float x[NS];
#pragma unroll
        for (int j = 0; j < NS; ++j) x[j] = bv * __expf(lb[j] - m);
        store_row12(&lds_x[l][0], x);
    }
    __syncthreads();

    for (int t = T_LEN - 2; t >= 0; --t) {
        // Row-owner: B_t row + 1/(scale_t+eps) in registers.
        float Bv[NS]; float inv_s = 0.0f;
        if (lane < 16) {
            const float* row = logB + ((size_t)(b0 + l) * T_LEN + t) * NS;
            float lb[NS];
            load_row12(row, lb);
            const float m = rowmax_shift(lb);
#pragma unroll
            for (int j = 0; j < NS; ++j) Bv[j] = __expf(lb[j] - m);
            inv_s = 1.0f / (ws_scale[(size_t)t * BATCH + b0 + l] + EPS_F);
            if (t > 0) __builtin_prefetch(row - NS, 0, 3);
        }

        // x_{t+1} @ P^T
        v8f acc = {};
#pragma unroll
        for (int c = 0; c < 3; ++c) {
            const int k = 4 * c + 2 * half;
            v2f a;
            a.x = lds_x[l][k];
            a.y = lds_x[l][k + 1];
            acc = __builtin_amdgcn_wmma_f32_16x16x4_f32(false, a, false, ptfrag[c],
                                                        (short)0, acc, false, false);
        }
        {
            float4* dst = reinterpret_cast<float4*>(&prodT[l][8 * half]);
            dst[0] = make_float4(acc[0], acc[1], acc[2], acc[3]);
            dst[1] = make_float4(acc[4], acc[5], acc[6], acc[7]);
        }
        __syncthreads();   // prodT visible; lds_x reads done before overwrite

        if (lane < 16) {
            float bvrow[NS];
#pragma unroll
            for (int j = 0; j < NS; ++j) bvrow[j] = prodT[j][l] * inv_s;
            store_row12(ws_beta + ((size_t)t * BATCH + b0 + l) * NS, bvrow);
            float x[NS];
#pragma unroll
            for (int j = 0; j < NS; ++j) x[j] = bvrow[j] * Bv[j];   // x_t for next step
            store_row12(&lds_x[l][0], x);
        }
        __syncthreads();
    }
}

// -------------------------------------------------------------------------
// Posteriors: gamma (all t) + xi (t < T-1). One wave32 per (b,t) pair;
// 262144 independent waves -> this is the bandwidth-bound, parallel part.
// -------------------------------------------------------------------------
__global__ __launch_bounds__(256)
void hmm_posteriors(const float* __restrict__ logB,
                    const float* __restrict__ P,
                    const float* __restrict__ ws_alpha,
                    const float* __restrict__ ws_beta,
                    float* __restrict__ out)
{
    const int  wave = threadIdx.x >> 5;
    const int  lane = threadIdx.x & 31;
    const long p    = (long)blockIdx.x * 8 + wave;
    if (p >= (long)BATCH * T_LEN) return;
    const int t = (int)(p % T_LEN);
    const int b = (int)(p / T_LEN);

    // ---- gamma[b,t,:] = alpha*beta / sum ----
    float av = 0.0f, bv = 0.0f;
    if (lane < NS) {
        av = ws_alpha[((size_t)t * BATCH + b) * NS + lane];
        bv = ws_beta [((size_t)t * BATCH + b) * NS + lane];
    }
    float g  = av * bv;
    float gs = g;
    gs += __shfl_xor(gs, 1);
    gs += __shfl_xor(gs, 2);
    gs += __shfl_xor(gs, 4);
    gs += __shfl_xor(gs, 8);
    gs += __shfl_xor(gs, 16);
    if (lane < NS)
        out[((size_t)b * T_LEN + t) * NS + lane] = g / gs;

    if (t >= T_LEN - 1) return;   // wave-uniform exit

    // ---- bprod_j = beta[t+1,b,j] * exp(logB[b,t+1,j] - min(rowmax,0)) ----
    float lb = (lane < NS)
             ? logB[((size_t)b * T_LEN + (t + 1)) * NS + lane]
             : -__builtin_inff();
    float m = lb;
    m = fmaxf(m, __shfl_xor(m, 1));
    m = fmaxf(m, __shfl_xor(m, 2));
    m = fmaxf(m, __shfl_xor(m, 4));
    m = fmaxf(m, __shfl_xor(m, 8));
    m = fmaxf(m, __shfl_xor(m, 16));
    m = fminf(m, 0.0f);
    float bp = 0.0f;
    if (lane < NS)
        bp = ws_beta[((size_t)(t + 1) * BATCH + b) * NS + lane] * __expf(lb - m);

    // ---- xi[b,t,i,j] = P[i,j]*alpha_i*bprod_j / sum ; 144 elems, 5/lane ----
    float val[5];
    float denom = 0.0f;
#pragma unroll
    for (int q = 0; q < 5; ++q) {
        const int e = lane + 32 * q;
        float v = 0.0f;
        if (e < NS * NS) {
            const int i = e / NS;
            const int j = e - i * NS;
            v = P[e] * __shfl(av, i) * __shfl(bp, j);
        }
        val[q] = v;
        denom += v;
    }
    denom += __shfl_xor(denom, 1);
    denom += __shfl_xor(denom, 2);
    denom += __shfl_xor(denom, 4);
    denom += __shfl_xor(denom, 8);
    denom += __shfl_xor(denom, 16);
    const float inv = 1.0f / denom;

    float* xo = out + (size_t)BATCH * T_LEN * NS
                    + ((size_t)b * (T_LEN - 1) + t) * (NS * NS);
#pragma unroll
    for (int q = 0; q < 5; ++q) {
        const int e = lane + 32 * q;
        if (e < NS * NS) xo[e] = val[q] * inv;
    }
}

// -------------------------------------------------------------------------
extern "C" void kernel_launch(void* const* d_in, const int* in_sizes, int n_in,
                              void* d_out, int out_size, void* d_ws, size_t ws_size,
                              hipStream_t stream)
{
    const float* logB = (const float*)d_in[0];   // (64, 4096, 12)
    const float* Pi0  = (const float*)d_in[1];   // (12,)
    const float* P    = (const float*)d_in[2];   // (12, 12)
    float* out = (float*)d_out;

    float* ws_alpha = (float*)d_ws;                              // T*B*NS
    float* ws_scale = ws_alpha + (size_t)T_LEN * BATCH * NS;     // T*B
    float* ws_beta  = ws_scale + (size_t)T_LEN * BATCH;          // T*B*NS

    hmm_forward_scan <<<dim3(BATCH / 16), dim3(32), 0, stream>>>(logB, Pi0, P, ws_alpha, ws_scale);
    hmm_backward_scan<<<dim3(BATCH / 16), dim3(32), 0, stream>>>(logB, P, ws_scale, ws_beta);

    const long pairs  = (long)BATCH * T_LEN;      // 262144 waves
    const int  blocks = (int)((pairs + 7) / 8);   // 8 waves / 256-thread block
    hmm_posteriors<<<dim3(blocks), dim3(256), 0, stream>>>(logB, P, ws_alpha, ws_beta, out);
}